// EdgeConv2_71124658422012
// MI455X (gfx1250) — compile-verified
//
#include <hip/hip_runtime.h>
#include <math.h>

// EdgeConv reference degenerates: neighbor gather is constant along the
// gathered axis, so the whole thing is a per-point 3-layer MLP (32->64->64->64)
// with inference BN + exact GELU, output transposed to [B, O, N].
//
// Memory-bound (~24 MB @ 23.3 TB/s ~= 1 us), so we keep full fp32 precision
// using V_WMMA_F32_16X16X4_F32. One wave = one 16-point tile through all
// three layers; inter-layer relayout via a per-wave LDS buffer (stride 66
// floats -> bank-conflict-free for both halves of the wave).

typedef __attribute__((ext_vector_type(2))) float v2f;
typedef __attribute__((ext_vector_type(8))) float v8f;

#define HSTR 66          // LDS row stride in floats (8*66 mod 64 == 16 -> halves on disjoint banks)
#define WPB  8           // waves per block

__device__ __forceinline__ v8f wmma4(v2f a, v2f b, v8f c) {
  // D(16x16,f32) = A(16x4,f32) * B(4x16,f32) + C
  return __builtin_amdgcn_wmma_f32_16x16x4_f32(
      /*neg_a=*/false, a, /*neg_b=*/false, b,
      /*c_mod=*/(short)0, c, /*reuse_a=*/false, /*reuse_b=*/false);
}

__device__ __forceinline__ float bn_gelu(float h, float sc, float bi) {
  float t = h * sc + bi;                       // folded inference BatchNorm
  return 0.5f * t * (1.0f + erff(t * 0.70710678118654752f));  // exact GELU
}

__global__ __launch_bounds__(256) void edgeconv_mlp_kernel(
    const float* __restrict__ x,   // [64,1024,32]
    const float* __restrict__ W1,  // [64,32]
    const float* __restrict__ g1, const float* __restrict__ b1,
    const float* __restrict__ m1, const float* __restrict__ v1,
    const float* __restrict__ W2,  // [64,64]
    const float* __restrict__ g2, const float* __restrict__ b2,
    const float* __restrict__ m2, const float* __restrict__ v2,
    const float* __restrict__ W3,  // [64,64]
    const float* __restrict__ g3, const float* __restrict__ b3,
    const float* __restrict__ m3, const float* __restrict__ v3,
    float* __restrict__ out)       // [64,64,1024]
{
  __shared__ float hbuf[WPB * 16 * HSTR];

  const int tid    = threadIdx.x;
  const int wave   = tid >> 5;
  const int lane   = tid & 31;
  const int lane_m = lane & 15;    // A-row / B-col / D-col index
  const int half   = lane >> 4;    // 0: K=kb+{0,1}, 1: K=kb+{2,3}
  const int kofs   = half * 2;

  const int tile = blockIdx.x * WPB + wave;   // 4096 tiles of 16 points
  const int p0   = tile * 16;
  float* H = hbuf + wave * 16 * HSTR;         // per-wave 16x64 activation tile

  // ---------------- Stage 1: h1 = bn_gelu(x @ W1^T), Kin = 32 ----------------
  v8f acc[4];
  const v8f vzero = {0.f, 0.f, 0.f, 0.f, 0.f, 0.f, 0.f, 0.f};
#pragma unroll
  for (int nt = 0; nt < 4; ++nt) acc[nt] = vzero;

  const float* arow = x + (size_t)(p0 + lane_m) * 32 + kofs;
#pragma unroll
  for (int kb = 0; kb < 32; kb += 4) {
    v2f a = *(const v2f*)(arow + kb);                       // A[m][kb+kofs..+1]
#pragma unroll
    for (int nt = 0; nt < 4; ++nt) {
      // B[k][o] = W1[o][k]  (row-major W1 -> contiguous float2 along k)
      v2f b = *(const v2f*)(W1 + (size_t)(nt * 16 + lane_m) * 32 + kb + kofs);
      acc[nt] = wmma4(a, b, acc[nt]);
    }
  }
#pragma unroll
  for (int nt = 0; nt < 4; ++nt) {
    int o = nt * 16 + lane_m;
    float sc = g1[o] * rsqrtf(v1[o] + 1e-5f);
    float bi = b1[o] - m1[o] * sc;
#pragma unroll
    for (int r = 0; r < 8; ++r) {
      int mrow = r + half * 8;                              // D-layout row
      H[mrow * HSTR + o] = bn_gelu(acc[nt][r], sc, bi);
    }
  }
  asm volatile("s_wait_dscnt 0" ::: "memory");

  // ---------------- Stages 2 & 3: Kin = 64, A from LDS ----------------
  const float* Ws[2] = {W2, W3};
  const float* Gs[2] = {g2, g3};
  const float* Bs[2] = {b2, b3};
  const float* Ms[2] = {m2, m3};
  const float* Vs[2] = {v2, v3};

#pragma unroll
  for (int s = 0; s < 2; ++s) {
    const float* W = Ws[s];
#pragma unroll
    for (int nt = 0; nt < 4; ++nt) acc[nt] = vzero;
#pragma unroll
    for (int kb = 0; kb < 64; kb += 4) {
      v2f a = *(const v2f*)(H + lane_m * HSTR + kb + kofs); // ds_load_b64
#pragma unroll
      for (int nt = 0; nt < 4; ++nt) {
        v2f b = *(const v2f*)(W + (size_t)(nt * 16 + lane_m) * 64 + kb + kofs);
        acc[nt] = wmma4(a, b, acc[nt]);
      }
    }
    asm volatile("s_wait_dscnt 0" ::: "memory");            // reads of H done before overwrite
#pragma unroll
    for (int nt = 0; nt < 4; ++nt) {
      int o = nt * 16 + lane_m;
      float sc = Gs[s][o] * rsqrtf(Vs[s][o] + 1e-5f);
      float bi = Bs[s][o] - Ms[s][o] * sc;
#pragma unroll
      for (int r = 0; r < 8; ++r) {
        int mrow = r + half * 8;
        H[mrow * HSTR + o] = bn_gelu(acc[nt][r], sc, bi);
      }
    }
    asm volatile("s_wait_dscnt 0" ::: "memory");
  }

  // ------------- Coalesced transpose-store: out[b][o][n] -------------
  // Lanes 0-15 write row o=2j, lanes 16-31 write row o=2j+1; 16 consecutive
  // n per half-wave -> contiguous 64B segments.
  const int bidx  = p0 >> 10;
  const int nbase = p0 & 1023;
  float* obase = out + (size_t)bidx * 64 * 1024 + nbase + lane_m;
#pragma unroll
  for (int j = 0; j < 32; ++j) {
    int o = 2 * j + half;
    obase[(size_t)o * 1024] = H[lane_m * HSTR + o];
  }
}

extern "C" void kernel_launch(void* const* d_in, const int* in_sizes, int n_in,
                              void* d_out, int out_size, void* d_ws, size_t ws_size,
                              hipStream_t stream) {
  const float* x  = (const float*)d_in[0];
  const float* W1 = (const float*)d_in[1];
  const float* g1 = (const float*)d_in[2];
  const float* b1 = (const float*)d_in[3];
  const float* m1 = (const float*)d_in[4];
  const float* v1 = (const float*)d_in[5];
  const float* W2 = (const float*)d_in[6];
  const float* g2 = (const float*)d_in[7];
  const float* b2 = (const float*)d_in[8];
  const float* m2 = (const float*)d_in[9];
  const float* v2 = (const float*)d_in[10];
  const float* W3 = (const float*)d_in[11];
  const float* g3 = (const float*)d_in[12];
  const float* b3 = (const float*)d_in[13];
  const float* m3 = (const float*)d_in[14];
  const float* v3 = (const float*)d_in[15];
  float* out = (float*)d_out;

  // P = 64*1024 = 65536 points -> 4096 tiles of 16 -> 8 waves/block -> 512 blocks
  edgeconv_mlp_kernel<<<512, 256, 0, stream>>>(
      x, W1, g1, b1, m1, v1, W2, g2, b2, m2, v2, W3, g3, b3, m3, v3, out);
}